// EGraphSAGE_48086453846628
// MI455X (gfx1250) — compile-verified
//
#include <hip/hip_runtime.h>
#include <hip/hip_bf16.h>

// ---------------------------------------------------------------------------
// EGraphSAGE forward, restructured via linearity of segment_sum:
//   segment_sum(x[src] @ W) == segment_sum(x[src]) @ W
// => 2 scatter-accumulate passes (f32 L2 atomics) + dense fp32 WMMA GEMMs
//    with persistent waves + TDM-staged weights + tiny edge gather.
// ---------------------------------------------------------------------------

typedef __attribute__((ext_vector_type(2))) float        v2f;
typedef __attribute__((ext_vector_type(8))) float        v8f;
typedef __attribute__((ext_vector_type(4))) unsigned int v4u;
typedef __attribute__((ext_vector_type(8))) int          v8i;
typedef __attribute__((ext_vector_type(4))) int          v4i;

#define N_NODES 50000
#define N_EDGES 800000
#define HID     64

#if defined(__HIP_DEVICE_COMPILE__) && \
    __has_builtin(__builtin_amdgcn_tensor_load_to_lds) && \
    __has_builtin(__builtin_amdgcn_s_wait_tensorcnt)
#define USE_TDM 1
#else
#define USE_TDM 0
#endif

#if defined(__HIP_DEVICE_COMPILE__) && \
    __has_builtin(__builtin_amdgcn_sched_group_barrier)
#define HAVE_SGB 1
#else
#define HAVE_SGB 0
#endif

// ---------------------------------------------------------------------------
// Scatter-accumulate. item < E: edge (src->dst) with edge attrs; item >= E:
// self-loop of node (item-E), zero attrs. 64 threads (2 waves) per item, one
// float each. unsafeAtomicAdd -> native global_atomic_add_f32 at L2.
// ---------------------------------------------------------------------------
__global__ __launch_bounds__(256) void scatter_accum(
    const float* __restrict__ X, const int* __restrict__ src,
    const int* __restrict__ dst, const float* __restrict__ EA,
    float* __restrict__ Ax, float* __restrict__ Ae, float* __restrict__ cnt,
    int E, int Ntot, int do_edge)
{
    int item = blockIdx.x * 4 + (threadIdx.x >> 6);
    int t    = threadIdx.x & 63;
    if (item >= Ntot) return;

    int s, d;
    if (item < E) { s = src[item]; d = dst[item]; }
    else          { s = d = item - E; }            // self loop

    unsafeAtomicAdd(&Ax[d * HID + t], X[s * HID + t]);

    if (do_edge) {
        if (item < E && t < 32)
            unsafeAtomicAdd(&Ae[d * 32 + t], EA[item * 32 + t]);
        if (t == 0)
            unsafeAtomicAdd(&cnt[d], 1.0f);
    }
}

#if USE_TDM
// ---------------------------------------------------------------------------
// Tensor Data Mover: stage a rows x 64 f32 row-major panel into LDS.
// Hand-packed 2D D# per CDNA5 ISA ch.8 (count=1, data_size=4B, tile==tensor,
// type=2). Groups 2/3 + extra group zero (2D tensor, no gather/iterate).
// This toolchain exposes the 6-arg builtin form.
// ---------------------------------------------------------------------------
__device__ __forceinline__ void tdm_load_panel(unsigned lds_byte_off,
                                               const float* gptr, int rows)
{
    unsigned long long ga = (unsigned long long)(uintptr_t)gptr;
    const int cols = HID;                                     // 64 floats/row
    v4u g0 = { 1u,                                            // count=1
               lds_byte_off,                                  // lds_addr
               (unsigned)(ga & 0xFFFFFFFFu),                  // global_addr lo
               (unsigned)((ga >> 32) & 0x01FFFFFFu) | (2u << 30) }; // hi|type=2
    v8i g1 = { (2 << 16),                                     // data_size=4B
               (cols & 0xFFFF) << 16,                         // tensor_dim0 lo
               ((cols >> 16) & 0xFFFF) | ((rows & 0xFFFF) << 16), // td0hi|td1lo
               ((rows >> 16) & 0xFFFF) | ((cols & 0xFFFF) << 16), // td1hi|tile0
               (rows & 0xFFFF),                               // tile_dim1 (tile2=0)
               cols,                                          // dim0_stride lo
               0, 0 };
    v4i z4 = { 0, 0, 0, 0 };
    v8i z8 = { 0, 0, 0, 0, 0, 0, 0, 0 };
    __builtin_amdgcn_tensor_load_to_lds(g0, g1, z4, z4, z8, 0);
}
#endif

// ---------------------------------------------------------------------------
// C[m,n] = A1[m,:]@W1 + A2[m,:]@W2, then (optional) /cnt[m], +bias, (opt) relu.
// One wave per 16x16 tile, V_WMMA_F32_16X16X4_F32 (exact fp32). Persistent
// waves: W1|W2 staged to LDS (TDM), all B fragments preloaded to registers.
// TWO independent accumulator chains (A1-chain, A2-chain) break the serial
// WMMA dependency so A-fragment loads of one chain overlap WMMAs of the
// other; sched_group_barrier requests a (4 loads | 4 WMMA) software pipeline.
// Fragment layouts per ISA 7.12.2:
//   A(16x4):  lane l, vgpr j -> A[l%16, 2*(l>>4)+j]
//   B(4x16):  lane l, vgpr j -> B[2*(l>>4)+j, l%16]
//   C(16x16): lane l, vgpr r -> C[r + 8*(l>>4), l%16]
// All branches are wave-uniform => EXEC all-ones around WMMA.
// ---------------------------------------------------------------------------
template <int K1, int K2, bool SCALE, bool RELU>
__global__ __launch_bounds__(128) void gemm2_wmma(
    const float* __restrict__ A1, const float* __restrict__ W1,
    const float* __restrict__ A2, const float* __restrict__ W2,
    const float* __restrict__ bias1, const float* __restrict__ bias2,
    const float* __restrict__ cnt, float* __restrict__ C, int Nrows)
{
    __shared__ float sW[(K1 + K2) * HID];

#if USE_TDM
    if (threadIdx.y == 0) {                  // wave 0 drives both TDM copies
        unsigned base = (unsigned)(uintptr_t)&sW[0];
        tdm_load_panel(base, W1, K1);
        tdm_load_panel(base + K1 * HID * 4, W2, K2);
        __builtin_amdgcn_s_wait_tensorcnt(0);
    }
    __syncthreads();
#else
    for (int i = threadIdx.y * 32 + threadIdx.x; i < (K1 + K2) * HID; i += 128)
        sW[i] = (i < K1 * HID) ? W1[i] : W2[i - K1 * HID];
    __syncthreads();
#endif

    const int lane = threadIdx.x;            // wave32: 0..31
    const int half = lane >> 4;
    const int lm   = lane & 15;
    const int n0   = threadIdx.y * 16;

    // Preload every B fragment (whole K range) into registers from LDS.
    v2f b1[K1 / 4], b2[K2 / 4];
#pragma unroll
    for (int k = 0; k < K1 / 4; ++k) {
        int row = 4 * k + 2 * half;
        b1[k].x = sW[row * HID + n0 + lm];
        b1[k].y = sW[(row + 1) * HID + n0 + lm];
    }
#pragma unroll
    for (int k = 0; k < K2 / 4; ++k) {
        int row = 4 * k + 2 * half;
        b2[k].x = sW[(K1 + row) * HID + n0 + lm];
        b2[k].y = sW[(K1 + row + 1) * HID + n0 + lm];
    }

    const float bias = bias1[n0 + lm] + (bias2 ? bias2[n0 + lm] : 0.0f);

    for (int m0 = blockIdx.x * 16; m0 < Nrows; m0 += gridDim.x * 16) {
        const v2f* a1 = (const v2f*)(A1 + (m0 + lm) * K1 + 2 * half);
        const v2f* a2 = (const v2f*)(A2 + (m0 + lm) * K2 + 2 * half);

        // Two independent WMMA chains -> scheduler can overlap loads of one
        // chain with matrix ops of the other.
        v8f acc0 = {0.f, 0.f, 0.f, 0.f, 0.f, 0.f, 0.f, 0.f};
        v8f acc1 = {0.f, 0.f, 0.f, 0.f, 0.f, 0.f, 0.f, 0.f};
#pragma unroll
        for (int k = 0; k < K1 / 4; ++k)
            acc0 = __builtin_amdgcn_wmma_f32_16x16x4_f32(
                false, a1[2 * k], false, b1[k], (short)0, acc0, false, false);
#pragma unroll
        for (int k = 0; k < K2 / 4; ++k)
            acc1 = __builtin_amdgcn_wmma_f32_16x16x4_f32(
                false, a2[2 * k], false, b2[k], (short)0, acc1, false, false);

#if HAVE_SGB
        // Request pipeline: (4 VMEM reads | 4 WMMA) repeated over both chains.
#pragma unroll
        for (int g = 0; g < (K1 + K2) / 16; ++g) {
            __builtin_amdgcn_sched_group_barrier(0x020, 4, 0); // VMEM read x4
            __builtin_amdgcn_sched_group_barrier(0x008, 4, 0); // WMMA x4
        }
#endif

#pragma unroll
        for (int r = 0; r < 8; ++r) {
            int   m = m0 + r + 8 * half;
            float v = acc0[r] + acc1[r];
            if (SCALE) v = v / cnt[m];       // mean (cnt >= 1: self loops)
            v += bias;
            if (RELU) v = fmaxf(v, 0.0f);
            C[m * HID + n0 + lm] = v;
        }
    }
}

// ---------------------------------------------------------------------------
// Per-node predictor projections: P[v,0:2] = z[v]@Wp[0:64,:],
//                                 P[v,2:4] = z[v]@Wp[64:128,:]
// ---------------------------------------------------------------------------
__global__ __launch_bounds__(256) void pred_node(
    const float* __restrict__ z, const float* __restrict__ Wp,
    float* __restrict__ P, int N)
{
    int v = blockIdx.x * 256 + threadIdx.x;
    if (v >= N) return;
    const float* zr = z + v * HID;
    float p0 = 0.f, p1 = 0.f, q0 = 0.f, q1 = 0.f;
#pragma unroll 8
    for (int k = 0; k < HID; ++k) {
        float zv = zr[k];
        p0 += zv * Wp[k * 2 + 0];
        p1 += zv * Wp[k * 2 + 1];
        q0 += zv * Wp[(HID + k) * 2 + 0];
        q1 += zv * Wp[(HID + k) * 2 + 1];
    }
    P[v * 4 + 0] = p0;
    P[v * 4 + 1] = p1;
    P[v * 4 + 2] = q0;
    P[v * 4 + 3] = q1;
}

// out[e] = P_top[row[e]] + P_bot[col[e]] + bp   over ORIGINAL edges
__global__ __launch_bounds__(256) void edge_out(
    const float* __restrict__ P, const int* __restrict__ src,
    const int* __restrict__ dst, const float* __restrict__ bp,
    float* __restrict__ out, int E)
{
    int e = blockIdx.x * 256 + threadIdx.x;
    if (e >= E) return;
    int r = src[e], c = dst[e];
    out[e * 2 + 0] = P[r * 4 + 0] + P[c * 4 + 2] + bp[0];
    out[e * 2 + 1] = P[r * 4 + 1] + P[c * 4 + 3] + bp[1];
}

extern "C" void kernel_launch(void* const* d_in, const int* in_sizes, int n_in,
                              void* d_out, int out_size, void* d_ws, size_t ws_size,
                              hipStream_t stream)
{
    const int N = N_NODES, E = N_EDGES, H = HID;

    const float* x   = (const float*)d_in[0];
    const int*   ei  = (const int*)d_in[1];
    const float* ea  = (const float*)d_in[2];
    const float* Wn1 = (const float*)d_in[3];  const float* bn1 = (const float*)d_in[4];
    const float* We1 = (const float*)d_in[5];  const float* be1 = (const float*)d_in[6];
    const float* Wu1 = (const float*)d_in[7];  const float* bu1 = (const float*)d_in[8];
    const float* Wn2 = (const float*)d_in[9];  const float* bn2 = (const float*)d_in[10];
    const float* We2 = (const float*)d_in[11]; const float* be2 = (const float*)d_in[12];
    const float* Wu2 = (const float*)d_in[13]; const float* bu2 = (const float*)d_in[14];
    const float* Wp  = (const float*)d_in[15]; const float* bp  = (const float*)d_in[16];

    const int* srcIdx = ei;         // edge_index[0]
    const int* dstIdx = ei + E;     // edge_index[1]
    float*     out    = (float*)d_out;

    // Workspace (floats): cnt[N] | Ax[64N] | Ae[32N] | aggr[64N] | h[64N]
    // Reuse: z -> Ax, P -> Ae (both free by the time they're overwritten).
    float* ws   = (float*)d_ws;
    float* cnt  = ws;
    float* Ax   = cnt  + N;
    float* Ae   = Ax   + (size_t)H * N;
    float* aggr = Ae   + (size_t)32 * N;
    float* h    = aggr + (size_t)H * N;
    float* z    = Ax;
    float* P    = Ae;

    const int  Ntot = E + N;                     // edges + self loops
    const dim3 sblk(256), sgrd((Ntot + 3) / 4);
    const dim3 gblk(32, H / 16), ggrd(512);      // persistent: ~6 tiles/wave

    // ---- conv1 ----
    (void)hipMemsetAsync(cnt, 0, sizeof(float) * (size_t)N * (1 + H + 32), stream);
    scatter_accum<<<sgrd, sblk, 0, stream>>>(x, srcIdx, dstIdx, ea, Ax, Ae, cnt,
                                             E, Ntot, 1);
    gemm2_wmma<64, 32, true, false><<<ggrd, gblk, 0, stream>>>(
        Ax, Wn1, Ae, We1, bn1, be1, cnt, aggr, N);
    gemm2_wmma<64, 64, false, true><<<ggrd, gblk, 0, stream>>>(
        x, Wu1, aggr, Wu1 + H * H, bu1, nullptr, nullptr, h, N);

    // ---- conv2 (Ae & cnt reused: edge attrs / degrees unchanged) ----
    (void)hipMemsetAsync(Ax, 0, sizeof(float) * (size_t)H * N, stream);
    scatter_accum<<<sgrd, sblk, 0, stream>>>(h, srcIdx, dstIdx, ea, Ax, nullptr,
                                             nullptr, E, Ntot, 0);
    gemm2_wmma<64, 32, true, false><<<ggrd, gblk, 0, stream>>>(
        Ax, Wn2, Ae, We2, bn2, be2, cnt, aggr, N);
    gemm2_wmma<64, 64, false, false><<<ggrd, gblk, 0, stream>>>(
        h, Wu2, aggr, Wu2 + H * H, bu2, nullptr, nullptr, z, N);

    // ---- predictor ----
    pred_node<<<(N + 255) / 256, 256, 0, stream>>>(z, Wp, P, N);
    edge_out<<<(E + 255) / 256, 256, 0, stream>>>(P, srcIdx, dstIdx, bp, out, E);
}